// RGCNLayer_60129542663
// MI455X (gfx1250) — compile-verified
//
#include <hip/hip_runtime.h>
#include <hip/hip_bf16.h>

// Problem constants (match reference)
#define NN    128   // nodes
#define TT    4096  // time/batch
#define NE    72    // edges
#define CIN   128   // IN
#define COUT  128   // OUT
#define NRULE 3

typedef float v2f __attribute__((ext_vector_type(2)));
typedef float v8f __attribute__((ext_vector_type(8)));

// D = A(16x4,f32) * B(4x16,f32) + C(16x16,f32)
// args: (neg_a, A, neg_b, B, c_mod, C, reuse_a, reuse_b)
__device__ __forceinline__ v8f wmma_f32(v2f a, v2f b, v8f c) {
  return __builtin_amdgcn_wmma_f32_16x16x4_f32(false, a, false, b, (short)0, c,
                                               false, false);
}

// ---------------------------------------------------------------------------
// Fused RGCN layer, gather formulation (no atomics):
//   out[n, t, :] = h_bias + feat[n, t, :] @ loop_weight
//                + sum_{e : dst[e]==n} sum_r truth[e,t,r] * (feat[src[e],t,:] @ W[et[e],r])
//
// grid = (T/32, N); block = 256 threads = 8 wave32s.
// Wave w owns OUT columns [16w, 16w+16); block covers a 32-row T slab
// (2 M-subtiles of 16 per wave). All matrix math in f32 WMMA 16x16x4.
// Edge scan is wave-uniform scalar control flow -> EXEC all-ones at WMMA.
//
// Fragment layouts (ISA 7.12.2, wave32):
//   A 16x4 : lane m=lane&15, a[v] = A[m][v + 2*(lane>>4)]
//   B 4x16 : b[v] = B[v + 2*(lane>>4)][lane&15]
//   C 16x16: c[v] at row v + 8*(lane>>4), col lane&15
// ---------------------------------------------------------------------------
__global__ __launch_bounds__(256) void rgcn_fused(
    const float* __restrict__ feat, const float* __restrict__ truth,
    const float* __restrict__ w, const float* __restrict__ lw,
    const float* __restrict__ bias, const int* __restrict__ src_idx,
    const int* __restrict__ dst_idx, const int* __restrict__ etypes,
    float* __restrict__ out) {
  const int n    = blockIdx.y;
  const int t0   = blockIdx.x * 32;
  const int wave = threadIdx.x >> 5;
  const int lane = threadIdx.x & 31;
  const int m    = lane & 15;  // A row / B,C col within a 16-tile
  const int kk   = lane >> 4;  // lane half
  const int n0   = wave * 16;

  // Output accumulators: 2 M-subtiles (rows t0..t0+15, t0+16..t0+31).
  v8f oacc0 = {}, oacc1 = {};

  // ---- Self-loop GEMM: feat[n] @ loop_weight ----
  {
    const float* ap0 = feat + ((size_t)n * TT + t0 + m) * CIN + 2 * kk;
    const float* ap1 = ap0 + 16 * CIN;
    const float* bp  = lw + (2 * kk) * COUT + n0 + m;
#pragma unroll 4
    for (int k = 0; k < CIN; k += 4) {
      v2f a0 = *(const v2f*)ap0;
      v2f a1 = *(const v2f*)ap1;
      v2f b;
      b.x = bp[0];
      b.y = bp[COUT];
      ap0 += 4;
      ap1 += 4;
      bp  += 4 * COUT;
      oacc0 = wmma_f32(a0, b, oacc0);
      oacc1 = wmma_f32(a1, b, oacc1);
    }
  }

  // ---- Gather incoming edges: uniform scalar scan over all 72 edges ----
  for (int e = 0; e < NE; ++e) {
    if (dst_idx[e] != n) continue;  // wave-uniform branch (SALU)

    const int src = src_idx[e];
    const int et  = etypes[e];

    v8f acc[2][NRULE] = {};

    const float* ap0 = feat + ((size_t)src * TT + t0 + m) * CIN + 2 * kk;
    const float* ap1 = ap0 + 16 * CIN;
    const float* bp0 =
        w + ((size_t)(et * NRULE + 0) * CIN + 2 * kk) * COUT + n0 + m;
    const float* bp1 = bp0 + (size_t)CIN * COUT;
    const float* bp2 = bp1 + (size_t)CIN * COUT;

#pragma unroll 4
    for (int k = 0; k < CIN; k += 4) {
      v2f a0 = *(const v2f*)ap0;
      v2f a1 = *(const v2f*)ap1;
      ap0 += 4;
      ap1 += 4;

      v2f b0, b1, b2;
      b0.x = bp0[0]; b0.y = bp0[COUT];
      b1.x = bp1[0]; b1.y = bp1[COUT];
      b2.x = bp2[0]; b2.y = bp2[COUT];
      bp0 += 4 * COUT;
      bp1 += 4 * COUT;
      bp2 += 4 * COUT;

      acc[0][0] = wmma_f32(a0, b0, acc[0][0]);
      acc[1][0] = wmma_f32(a1, b0, acc[1][0]);
      acc[0][1] = wmma_f32(a0, b1, acc[0][1]);
      acc[1][1] = wmma_f32(a1, b1, acc[1][1]);
      acc[0][2] = wmma_f32(a0, b2, acc[0][2]);
      acc[1][2] = wmma_f32(a1, b2, acc[1][2]);
    }

    // Mix the 3 rule accumulators by per-row truth values; fold into oacc.
    const float* tvb = truth + (size_t)e * TT * NRULE;
#pragma unroll
    for (int v = 0; v < 8; ++v) {
      const int t = t0 + v + 8 * kk;
      const float* tv = tvb + (size_t)t * NRULE;
      oacc0[v] +=
          tv[0] * acc[0][0][v] + tv[1] * acc[0][1][v] + tv[2] * acc[0][2][v];
      const float* tv1 = tv + 16 * NRULE;
      oacc1[v] += tv1[0] * acc[1][0][v] + tv1[1] * acc[1][1][v] +
                  tv1[2] * acc[1][2][v];
    }
  }

  // ---- Bias + single coalesced store pass ----
  const float bv = bias[n0 + m];
  float* ob = out + ((size_t)n * TT + t0) * COUT + n0 + m;
#pragma unroll
  for (int v = 0; v < 8; ++v) {
    const int row = v + 8 * kk;
    ob[(size_t)row * COUT]        = oacc0[v] + bv;
    ob[(size_t)(row + 16) * COUT] = oacc1[v] + bv;
  }
}

extern "C" void kernel_launch(void* const* d_in, const int* in_sizes, int n_in,
                              void* d_out, int out_size, void* d_ws,
                              size_t ws_size, hipStream_t stream) {
  const float* feat   = (const float*)d_in[0];  // [N,T,IN]
  const float* truth  = (const float*)d_in[1];  // [E,T,NRULE]
  const float* weight = (const float*)d_in[2];  // [NREL,NRULE,IN,OUT]
  const float* lw     = (const float*)d_in[3];  // [IN,OUT]
  const float* bias   = (const float*)d_in[4];  // [OUT]
  const int* src_idx  = (const int*)d_in[5];    // [E]
  const int* dst_idx  = (const int*)d_in[6];    // [E]
  const int* etypes   = (const int*)d_in[7];    // [E]
  float* out = (float*)d_out;                   // [N,T,OUT]

  rgcn_fused<<<dim3(TT / 32, NN), dim3(256), 0, stream>>>(
      feat, truth, weight, lw, bias, src_idx, dst_idx, etypes, out);
}